// HyperGCN_35519379538265
// MI455X (gfx1250) — compile-verified
//
#include <hip/hip_runtime.h>
#include <hip/hip_bf16.h>
#include <math.h>

// ---------------------------------------------------------------------------
// M3NET hyper-GCN forward for MI455X (gfx1250, wave32, WMMA + TDM).
// D_MODEL = 512 fixed (reference constant).
// ---------------------------------------------------------------------------

#define DM 512

typedef __attribute__((ext_vector_type(2))) float v2f;
typedef __attribute__((ext_vector_type(8))) float v8f;
typedef __attribute__((ext_vector_type(4))) unsigned int v4u;
typedef __attribute__((ext_vector_type(8))) int v8i;
typedef __attribute__((ext_vector_type(4))) int v4i;

#if defined(__has_builtin)
#if __has_builtin(__builtin_amdgcn_wmma_f32_16x16x4_f32)
#define HAVE_WMMA_F32X4 1
#endif
#if __has_builtin(__builtin_amdgcn_tensor_load_to_lds) && \
    __has_builtin(__builtin_amdgcn_s_wait_tensorcnt)
#define HAVE_TDM 1
#endif
#endif

// ---------------------------------------------------------------------------
// Index construction (single block). Reconstructs the reference's
// create_hyper_index / create_gnn_index structures from dia_len on device.
// hdr[0]=num_nodes hdr[1]=num_edges hdr[2]=gnn_edge_count hdr[3]=N_utt
// ---------------------------------------------------------------------------
__global__ __launch_bounds__(256) void build_idx_kernel(
    const int* __restrict__ dia_len, const float* __restrict__ qmask,
    int ND, int* hdr, int* dia_t, int* goff, int* spk,
    int* n_idx, int* e_idx, int* feat_idx,
    int* gsrc, int* gdst, int* lord, int* aord, int* vord)
{
  const int tid = threadIdx.x;
  if (tid == 0) {
    int t = 0, g = 0;
    for (int d = 0; d < ND; ++d) {
      dia_t[d] = t; goff[d] = g;
      int i = dia_len[d];
      t += i;
      g += 3 * i * (i - 1) + 6 * i;
    }
    dia_t[ND] = t; goff[ND] = g;
    hdr[0] = 3 * t;            // num_nodes
    hdr[1] = t + 3 * ND;       // num_edges
    hdr[2] = g + 3 * t;        // gnn edges (+ self loops)
    hdr[3] = t;                // N_utt
  }
  __syncthreads();
  const int NUtt = dia_t[ND];
  for (int d = 0; d < ND; ++d) {
    const int t  = dia_t[d];
    const int i  = dia_len[d];
    const int nb = 3 * t;          // node base
    const int eb = t + 3 * d;      // hyperedge base
    const int pb = 6 * t;          // incidence-pair base
    const int gb = goff[d];        // gnn edge base
    for (int p = tid; p < i; p += blockDim.x) {
      int u = t + p;
      float q0 = qmask[(p * ND + d) * 2 + 0];
      float q1 = qmask[(p * ND + d) * 2 + 1];
      spk[u] = (q1 > q0) ? 1 : 0;
      lord[u] = nb + p;
      aord[u] = nb + i + p;
      vord[u] = nb + 2 * i + p;
      feat_idx[nb + p]         = u;             // l
      feat_idx[nb + i + p]     = u + NUtt;      // a
      feat_idx[nb + 2 * i + p] = u + 2 * NUtt;  // v
      for (int m = 0; m < 3; ++m) {
        n_idx[pb + m * i + p] = nb + m * i + p;   // modality hyperedges
        e_idx[pb + m * i + p] = eb + m;
        n_idx[pb + 3 * i + 3 * p + m] = nb + m * i + p;  // trio hyperedges
        e_idx[pb + 3 * i + 3 * p + m] = eb + 3 + p;
      }
    }
    const int ce = i * (i - 1);  // directed clique edges per modality
    for (int idx = tid; idx < 3 * ce; idx += blockDim.x) {
      int m = idx / ce, r = idx % ce;
      int a_ = r / (i - 1), b_ = r % (i - 1);
      if (b_ >= a_) b_ += 1;
      gsrc[gb + idx] = nb + m * i + a_;
      gdst[gb + idx] = nb + m * i + b_;
    }
    const int pq_[6][2] = {{0,1},{0,2},{1,0},{1,2},{2,0},{2,1}};
    for (int idx = tid; idx < 6 * i; idx += blockDim.x) {
      int e = idx / i, u2 = idx % i;
      gsrc[gb + 3 * ce + idx] = nb + pq_[e][0] * i + u2;
      gdst[gb + 3 * ce + idx] = nb + pq_[e][1] * i + u2;
    }
  }
  __syncthreads();
  const int NNodes = hdr[0];
  const int gbase  = goff[ND];
  for (int n = tid; n < NNodes; n += blockDim.x) {
    gsrc[gbase + n] = n;
    gdst[gbase + n] = n;
  }
}

// Dv[n] += he_w[e], Be[e] += EW_w[n] over incidence pairs
__global__ void dvbe_kernel(const int* __restrict__ n_idx, const int* __restrict__ e_idx,
                            const float* __restrict__ hew, const float* __restrict__ eww,
                            float* Dv, float* Be, int NP)
{
  int j = blockIdx.x * blockDim.x + threadIdx.x;
  if (j >= NP) return;
  atomicAdd(&Dv[n_idx[j]], hew[e_idx[j]]);
  atomicAdd(&Be[e_idx[j]], eww[n_idx[j]]);
}

__global__ void deg_kernel(const int* __restrict__ hdr, const int* __restrict__ gsrc, float* deg)
{
  int gcnt = hdr[2];
  for (int j = blockIdx.x * blockDim.x + threadIdx.x; j < gcnt;
       j += gridDim.x * blockDim.x)
    atomicAdd(&deg[gsrc[j]], 1.0f);
}

__global__ void invert_kernel(float* Dv, float* Be, const float* deg, float* dis, int NN, int NE)
{
  int j = blockIdx.x * blockDim.x + threadIdx.x;
  if (j < NN) {
    float v = Dv[j]; Dv[j] = (v > 0.f) ? 1.f / v : 0.f;
    float d = deg[j]; dis[j] = (d > 0.f) ? rsqrtf(d) : 0.f;
  }
  if (j < NE) {
    float v = Be[j]; Be[j] = (v > 0.f) ? 1.f / v : 0.f;
  }
}

// features[node][c] = stacked[feat_idx[node]][c]  (+ spk_emb for l rows)
__global__ void prep_kernel(const float* __restrict__ a, const float* __restrict__ v,
                            const float* __restrict__ l, const float* __restrict__ spk_emb,
                            const int* __restrict__ spk, const int* __restrict__ feat_idx,
                            float* feats, int NN, int NU)
{
  int t = blockIdx.x * blockDim.x + threadIdx.x;
  if (t >= NN * DM) return;
  int r = t >> 9, c = t & (DM - 1);
  int fi = feat_idx[r];
  float val;
  if (fi < NU)            val = l[(size_t)fi * DM + c] + spk_emb[(size_t)spk[fi] * DM + c];
  else if (fi < 2 * NU)   val = a[(size_t)(fi - NU) * DM + c];
  else                    val = v[(size_t)(fi - 2 * NU) * DM + c];
  feats[t] = val;
}

// ---------------------------------------------------------------------------
// fp32 WMMA GEMM:  C[M x 512] = A[M x 512] @ W[512 x 512] (+ bias)
// grid = (16, ceil(Mtiles/8)), block = 256 (8 wave32).
// The 512x32 W panel (64 KB) is DMA'd into LDS by the Tensor Data Mover
// (one TENSOR_LOAD_TO_LDS issued by wave 0, tracked by TENSORcnt), stored
// row-major sW[k][c]. Each wave owns a 16x32 output tile -> two v8f
// accumulators, 256 v_wmma_f32_16x16x4_f32 per wave.
// Fragment layouts per CDNA5 ISA 7.12.2 (32-bit A 16x4 and C/D 16x16).
//
// TDM aliasing note (rounds 3-5): the DMA writes LDS out-of-band. A bare
// memory-clobber asm is NOT enough (non-escaped object => clobber can't
// alias it => loads folded to undef), and volatile reads forced serialized
// flat SYS loads. The working recipe: escape &sW[0] as an *operand* of the
// memory-clobber asm (object escapes, clobber may write it), then read via
// plain sW[...] so infer-address-spaces keeps the accesses on the ds path.
// ---------------------------------------------------------------------------
__global__ __launch_bounds__(256) void wmma_gemm_kernel(
    const float* __restrict__ A, const float* __restrict__ W,
    const float* __restrict__ bias, float* __restrict__ C, int M)
{
  __shared__ float sW[DM * 32];   // sole LDS object -> LDS byte offset 0
  const int tid = threadIdx.x;
  const int n0  = blockIdx.x * 32;

#ifdef HAVE_TDM
  if (tid < 32) {  // wave 0 issues the tensor DMA (EXEC ignored by TDM)
    unsigned long long ga = (unsigned long long)(const void*)(W + n0);
    // D# group 0: count=1, lds_addr=0, global tile address, type=2 (image)
    v4u g0 = { 1u, 0u, (unsigned)ga,
               (unsigned)((ga >> 32) & 0x01FFFFFFu) | 0x80000000u };
    // D# group 1 (ISA 8.4): data_size=4B(code 2); tensor 32x512 (dim0 x dim1),
    // tile 32x512, row stride 512 elements.
    v8i g1 = { (int)(2u << 16),     // [17:16] data_size
               (int)(32u << 16),    // [79:48] tensor_dim0 = 32 (low half)
               (int)(512u << 16),   // [111:80] tensor_dim1 = 512 (low half)
               (int)(32u << 16),    // [127:112] tile_dim0 = 32
               (int)512,            // [143:128] tile_dim1 = 512, tile_dim2 = 0
               (int)512,            // [207:160] tensor_dim0_stride = 512
               (int)(512u << 16),   // [255:208] tensor_dim1_stride = 512
               0 };
    v4i gz4 = { 0, 0, 0, 0 };           // <=2D tensor: groups 2/3 unused
    v8i gz8 = { 0, 0, 0, 0, 0, 0, 0, 0 };
    // clang-23 / therock-10.0 six-operand form: (g0, g1, g2, g3, gX, cpol)
    __builtin_amdgcn_tensor_load_to_lds(g0, g1, gz4, gz4, gz8, 0);
    __builtin_amdgcn_s_wait_tensorcnt(0);
  }
  __syncthreads();
  // Escape the LDS object into the clobber so it may alias the asm's writes:
  // this keeps the sW loads alive without volatile (which forced flat SYS).
  asm volatile("" : : "v"(&sW[0]) : "memory");
#else
  for (int idx = tid; idx < DM * 32; idx += 256) {
    int k = idx >> 5, c = idx & 31;
    sW[k * 32 + c] = W[(size_t)k * DM + n0 + c];
  }
  __syncthreads();
#endif

  const int wave  = tid >> 5;
  const int lane  = tid & 31;
  const int mtile = blockIdx.y * 8 + wave;
  const int m0    = mtile * 16;
  if (m0 >= M) return;

  const int lrow  = lane & 15;   // A row within tile / B,C column within tile
  const int khalf = lane >> 4;   // 0: K = k0,k0+1   1: K = k0+2,k0+3
  int arow = m0 + lrow;
  if (arow >= M) arow = M - 1;   // clamp reads in partial tile (stores guarded)
  const float* Aptr = A + (size_t)arow * DM + khalf * 2;

  v8f acc0 = {0.f, 0.f, 0.f, 0.f, 0.f, 0.f, 0.f, 0.f};
  v8f acc1 = acc0;
  for (int k0 = 0; k0 < DM; k0 += 4) {
    __builtin_prefetch(Aptr + k0 + 64, 0, 0);
    v2f a = *(const v2f*)(Aptr + k0);
    const int kk = k0 + khalf * 2;   // even
    const float* bp = &sW[kk * 32];
    v2f b0 = { bp[lrow],      bp[32 + lrow] };
    v2f b1 = { bp[lrow + 16], bp[48 + lrow] };
#ifdef HAVE_WMMA_F32X4
    acc0 = __builtin_amdgcn_wmma_f32_16x16x4_f32(false, a, false, b0, (short)0, acc0, false, false);
    acc1 = __builtin_amdgcn_wmma_f32_16x16x4_f32(false, a, false, b1, (short)0, acc1, false, false);
#else
    acc0[0] += a.x * b0.x; acc1[0] += a.x * b1.x;  // keep TU compiling
#endif
  }
  // C/D layout: VGPR r -> row m0+r (lanes 0-15) / m0+8+r (lanes 16-31)
  const int rbase = m0 + khalf * 8;
  const int c0 = n0 + lrow, c1 = c0 + 16;
  const float bv0 = bias ? bias[c0] : 0.f;
  const float bv1 = bias ? bias[c1] : 0.f;
#pragma unroll
  for (int r = 0; r < 8; ++r) {
    int row = rbase + r;
    if (row < M) {
      C[(size_t)row * DM + c0] = acc0[r] + bv0;
      C[(size_t)row * DM + c1] = acc1[r] + bv1;
    }
  }
}

// hyperconv scatter 1: edge_feat[e] += xl[n] over incidence pairs
__global__ void hscat1_kernel(const int* __restrict__ n_idx, const int* __restrict__ e_idx,
                              const float* __restrict__ xl, float* ef, int NP)
{
  int t = blockIdx.x * blockDim.x + threadIdx.x;
  if (t >= NP * DM) return;
  int j = t >> 9, c = t & (DM - 1);
  atomicAdd(&ef[(size_t)e_idx[j] * DM + c], xl[(size_t)n_idx[j] * DM + c]);
}

// hyperconv scatter 2: nacc[n] += Be[e] * edge_feat[e]
__global__ void hscat2_kernel(const int* __restrict__ n_idx, const int* __restrict__ e_idx,
                              const float* __restrict__ Be, const float* __restrict__ ef,
                              float* nacc, int NP)
{
  int t = blockIdx.x * blockDim.x + threadIdx.x;
  if (t >= NP * DM) return;
  int j = t >> 9, c = t & (DM - 1);
  int e = e_idx[j];
  atomicAdd(&nacc[(size_t)n_idx[j] * DM + c], Be[e] * ef[(size_t)e * DM + c]);
}

__global__ void hfinal_kernel(const float* __restrict__ Dv, const float* __restrict__ nacc,
                              const float* __restrict__ hb, float* out, int NN)
{
  int t = blockIdx.x * blockDim.x + threadIdx.x;
  if (t >= NN * DM) return;
  int n = t >> 9, c = t & (DM - 1);
  out[t] = Dv[n] * nacc[t] + hb[c];
}

// per-node dot products against the two halves of gate_w (wave32 reductions)
__global__ __launch_bounds__(256) void pq_kernel(
    const float* __restrict__ xl, const float* __restrict__ gw,
    float* p, float* q, int NN)
{
  const int wave = threadIdx.x >> 5, lane = threadIdx.x & 31;
  const int n = blockIdx.x * 8 + wave;
  if (n >= NN) return;
  const float* x = xl + (size_t)n * DM;
  float sp = 0.f, sq = 0.f;
  for (int c = lane; c < DM; c += 32) {
    float xv = x[c];
    sp += xv * gw[c];
    sq += xv * gw[DM + c];
  }
  for (int o = 16; o; o >>= 1) {
    sp += __shfl_xor(sp, o, 32);
    sq += __shfl_xor(sq, o, 32);
  }
  if (lane == 0) { p[n] = sp; q[n] = sq; }
}

// coef[e] = dis[src]*dis[dst]*tanh(p[dst] + q[src] + gate_b)
__global__ void coef_kernel(const int* __restrict__ hdr, const int* __restrict__ gsrc,
                            const int* __restrict__ gdst, const float* __restrict__ p,
                            const float* __restrict__ q, const float* __restrict__ dis,
                            const float* __restrict__ gb, int kk, float* coef)
{
  int gcnt = hdr[2];
  for (int j = blockIdx.x * blockDim.x + threadIdx.x; j < gcnt;
       j += gridDim.x * blockDim.x) {
    int s = gsrc[j], d = gdst[j];
    coef[j] = dis[s] * dis[d] * tanhf(p[d] + q[s] + gb[kk]);
  }
}

// gacc[dst] += coef[e] * xl[src]
__global__ void msg_kernel(const int* __restrict__ hdr, const int* __restrict__ gsrc,
                           const int* __restrict__ gdst, const float* __restrict__ coef,
                           const float* __restrict__ xl, float* gacc)
{
  long gcnt  = hdr[2];
  long total = gcnt << 9;
  for (long t = (long)blockIdx.x * blockDim.x + threadIdx.x; t < total;
       t += (long)gridDim.x * blockDim.x) {
    int j = (int)(t >> 9), c = (int)(t & (DM - 1));
    atomicAdd(&gacc[(size_t)gdst[j] * DM + c], coef[j] * xl[(size_t)gsrc[j] * DM + c]);
  }
}

__global__ void gupd_kernel(float* g, const float* __restrict__ gacc, int NN)
{
  int t = blockIdx.x * blockDim.x + threadIdx.x;
  if (t < NN * DM) g[t] += gacc[t];
}

// final: out[u] = [out2[lord[u]], out2[aord[u]], out2[vord[u]]],
//        out2[n] = [features[n], hyper_out[n], g[n]]
__global__ void concat_kernel(const float* __restrict__ feats, const float* __restrict__ outh,
                              const float* __restrict__ g, const int* __restrict__ lord,
                              const int* __restrict__ aord, const int* __restrict__ vord,
                              float* __restrict__ out, int NU)
{
  int t = blockIdx.x * blockDim.x + threadIdx.x;
  if (t >= NU * 9 * DM) return;
  int r = t / (9 * DM);
  int j = t - r * (9 * DM);
  int sel = j / (3 * DM), w = j - sel * (3 * DM);
  int node = (sel == 0) ? lord[r] : ((sel == 1) ? aord[r] : vord[r]);
  int part = w / DM, c = w - part * DM;
  const float* src = (part == 0) ? feats : ((part == 1) ? outh : g);
  out[t] = src[(size_t)node * DM + c];
}

// ---------------------------------------------------------------------------
// Host driver
// ---------------------------------------------------------------------------
extern "C" void kernel_launch(void* const* d_in, const int* in_sizes, int n_in,
                              void* d_out, int out_size, void* d_ws, size_t ws_size,
                              hipStream_t stream)
{
  const float* a_in   = (const float*)d_in[0];
  const float* v_in   = (const float*)d_in[1];
  const float* l_in   = (const float*)d_in[2];
  const float* qmask  = (const float*)d_in[3];
  const int*   dialen = (const int*)  d_in[4];
  const float* spkemb = (const float*)d_in[5];
  const float* fc1_w  = (const float*)d_in[6];
  const float* fc1_b  = (const float*)d_in[7];
  const float* he_w   = (const float*)d_in[8];
  const float* ew_w   = (const float*)d_in[9];
  const float* hyp_w  = (const float*)d_in[10];
  const float* hyp_b  = (const float*)d_in[11];
  const float* gcn_w  = (const float*)d_in[12];
  const float* gcn_b  = (const float*)d_in[13];
  const float* gate_w = (const float*)d_in[14];
  const float* gate_b = (const float*)d_in[15];

  const int NU   = in_sizes[0] / DM;           // utterances
  const int ND   = in_sizes[4];                // dialogs
  const int DLEN = in_sizes[3] / (ND * 2);     // max dialog length
  const int NN   = 3 * NU;                     // nodes
  const int NE   = NU + 3 * ND;                // hyperedges
  const int NP   = 6 * NU;                     // incidence pairs
  const int GMAX = ND * (3 * DLEN * (DLEN - 1) + 6 * DLEN) + NN;  // gnn edge bound

  // ---- workspace carve-out (256B-aligned slices) ----
  char* w = (char*)d_ws;
  size_t off = 0;
  auto alloc = [&](size_t elems) -> void* {
    void* p = w + off;
    off += ((elems * 4) + 255) & ~(size_t)255;
    return p;
  };
  int* hdr      = (int*)alloc(64);
  int* dia_t    = (int*)alloc(ND + 1);
  int* goff     = (int*)alloc(ND + 1);
  int* spk      = (int*)alloc(NU);
  int* n_idx    = (int*)alloc(NP);
  int* e_idx    = (int*)alloc(NP);
  int* feat_idx = (int*)alloc(NN);
  int* lord     = (int*)alloc(NU);
  int* aord     = (int*)alloc(NU);
  int* vord     = (int*)alloc(NU);
  int* gsrc     = (int*)alloc(GMAX);
  int* gdst     = (int*)alloc(GMAX);
  float* Dv     = (float*)alloc(NN);
  float* Be     = (float*)alloc(NE);
  float* deg    = (float*)alloc(NN);
  float* dis    = (float*)alloc(NN);
  float* pbuf   = (float*)alloc(NN);
  float* qbuf   = (float*)alloc(NN);
  float* coef   = (float*)alloc(GMAX);
  float* feats  = (float*)alloc((size_t)NN * DM);
  float* x1     = (float*)alloc((size_t)NN * DM);
  float* outh   = (float*)alloc((size_t)NN * DM);
  float* gbuf   = (float*)alloc((size_t)NN * DM);
  float* xl     = (float*)alloc((size_t)NN * DM);
  float* ef     = (float*)alloc((size_t)NE * DM);
  float* nacc   = (float*)alloc((size_t)NN * DM);
  float* gacc   = (float*)alloc((size_t)NN * DM);
  (void)ws_size; (void)n_in; (void)out_size;

  const int B = 256;
  const int mtiles = (NN + 15) / 16;
  dim3 ggrid(DM / 32, (mtiles + 7) / 8);

  // 1. indices + speaker ids
  build_idx_kernel<<<1, B, 0, stream>>>(dialen, qmask, ND, hdr, dia_t, goff, spk,
                                        n_idx, e_idx, feat_idx, gsrc, gdst,
                                        lord, aord, vord);
  // 2. normalizations
  (void)hipMemsetAsync(Dv, 0, (size_t)NN * 4, stream);
  (void)hipMemsetAsync(Be, 0, (size_t)NE * 4, stream);
  (void)hipMemsetAsync(deg, 0, (size_t)NN * 4, stream);
  dvbe_kernel<<<(NP + B - 1) / B, B, 0, stream>>>(n_idx, e_idx, he_w, ew_w, Dv, Be, NP);
  deg_kernel<<<(GMAX + B - 1) / B, B, 0, stream>>>(hdr, gsrc, deg);
  invert_kernel<<<(NN + B - 1) / B, B, 0, stream>>>(Dv, Be, deg, dis, NN, NE);

  // 3. features (permuted + speaker embedding) and x1 = features @ fc1 + b
  prep_kernel<<<((size_t)NN * DM + B - 1) / B, B, 0, stream>>>(
      a_in, v_in, l_in, spkemb, spk, feat_idx, feats, NN, NU);
  wmma_gemm_kernel<<<ggrid, B, 0, stream>>>(feats, fc1_w, fc1_b, x1, NN);

  // g starts at x1
  (void)hipMemcpyAsync(gbuf, x1, (size_t)NN * DM * 4, hipMemcpyDeviceToDevice, stream);

  // 4. three hypergraph conv layers
  const float* hin = x1;
  for (int ll = 0; ll < 3; ++ll) {
    wmma_gemm_kernel<<<ggrid, B, 0, stream>>>(hin, hyp_w + (size_t)ll * DM * DM, nullptr, xl, NN);
    (void)hipMemsetAsync(ef, 0, (size_t)NE * DM * 4, stream);
    hscat1_kernel<<<((size_t)NP * DM + B - 1) / B, B, 0, stream>>>(n_idx, e_idx, xl, ef, NP);
    (void)hipMemsetAsync(nacc, 0, (size_t)NN * DM * 4, stream);
    hscat2_kernel<<<((size_t)NP * DM + B - 1) / B, B, 0, stream>>>(n_idx, e_idx, Be, ef, nacc, NP);
    hfinal_kernel<<<((size_t)NN * DM + B - 1) / B, B, 0, stream>>>(Dv, nacc, hyp_b + (size_t)ll * DM, outh, NN);
    hin = outh;
  }

  // 5. four gated GCN layers
  for (int kk = 0; kk < 4; ++kk) {
    wmma_gemm_kernel<<<ggrid, B, 0, stream>>>(gbuf, gcn_w + (size_t)kk * DM * DM,
                                              gcn_b + (size_t)kk * DM, xl, NN);
    pq_kernel<<<(NN + 7) / 8, B, 0, stream>>>(xl, gate_w + (size_t)kk * 2 * DM, pbuf, qbuf, NN);
    coef_kernel<<<(GMAX + B - 1) / B, B, 0, stream>>>(hdr, gsrc, gdst, pbuf, qbuf, dis, gate_b, kk, coef);
    (void)hipMemsetAsync(gacc, 0, (size_t)NN * DM * 4, stream);
    msg_kernel<<<((size_t)GMAX * DM + B - 1) / B, B, 0, stream>>>(hdr, gsrc, gdst, coef, xl, gacc);
    gupd_kernel<<<((size_t)NN * DM + B - 1) / B, B, 0, stream>>>(gbuf, gacc, NN);
  }

  // 6. final modality-concat output (840 x 4608)
  concat_kernel<<<((size_t)NU * 9 * DM + B - 1) / B, B, 0, stream>>>(
      feats, outh, gbuf, lord, aord, vord, (float*)d_out, NU);
}